// BiInteraction_22874995819092
// MI455X (gfx1250) — compile-verified
//
#include <hip/hip_runtime.h>
#include <hip/hip_bf16.h>

typedef float v2f __attribute__((ext_vector_type(2)));
typedef float v8f __attribute__((ext_vector_type(8)));

#define B_    64
#define L_    1000
#define N_    2560
#define D_    128
#define NPB   40          // atoms per molecule (N/B)
#define H1_   512
#define H2_   256
#define NEGV  (-9.0e15f)
#define QS_STRIDE 132     // 128 + 4 pad -> avoid 16-way LDS bank conflicts
#define NT_TILES  63      // ceil(1000/16)

// -------- float atomic max on LDS via CAS --------
__device__ __forceinline__ void atomicMaxF(float* addr, float val) {
    unsigned int* ua = (unsigned int*)addr;
    unsigned int cur = __float_as_uint(*addr);
    while (__uint_as_float(cur) < val) {
        unsigned int prev = atomicCAS(ua, cur, __float_as_uint(val));
        if (prev == cur) break;
        cur = prev;
    }
}

// generic pointer to a __shared__ object: low 32 bits are the LDS byte offset
__device__ __forceinline__ unsigned lds_off(const void* p) {
    return (unsigned)(size_t)p;
}

// ============================================================
// Kernel 1: q = atom_embed @ Wa   [2560,128] x [128,128]
// one wave per 16x16 output tile, K=128 in 32 wmma_f32_16x16x4 steps
// ============================================================
__global__ void __launch_bounds__(256)
k_q_gemm(const float* __restrict__ atom, const float* __restrict__ Wa,
         float* __restrict__ q) {
    const int lane = threadIdx.x & 31;
    const int wv   = threadIdx.x >> 5;     // 0..7 -> n-tile (8*16 = 128 cols)
    const int mt   = blockIdx.x;           // 0..159
    const int col  = lane & 15;
    const int half = lane >> 4;
    const int nb   = wv * 16;

    v8f c = {};
    const float* arow = atom + (size_t)(mt * 16 + col) * D_;   // A row = lane&15
    #pragma unroll 4
    for (int ks = 0; ks < D_ / 4; ++ks) {
        const int kb = ks * 4 + half * 2;                      // K base for this half
        v2f a = *(const v2f*)(arow + kb);                      // A[m, kb..kb+1]
        v2f b;
        b.x = Wa[(size_t)kb * D_ + nb + col];                  // B[kb,   n]
        b.y = Wa[(size_t)(kb + 1) * D_ + nb + col];            // B[kb+1, n]
        c = __builtin_amdgcn_wmma_f32_16x16x4_f32(false, a, false, b,
                                                  (short)0, c, false, false);
    }
    // C/D layout: vgpr r -> M = r + 8*half, lane&15 -> N
    float* qo = q + (size_t)(mt * 16 + half * 8) * D_ + nb + col;
    #pragma unroll
    for (int r = 0; r < 8; ++r) qo[(size_t)r * D_] = c[r];
}

// ============================================================
// Kernel 2: per-molecule fused bilinear attention
//   S = tanh(Q_b @ P_b^T) (masked), row/col reductions, both pools -> h0[b,256]
//   P tiles staged to LDS via async-to-LDS (ASYNCcnt), WMMA fed from LDS
// ============================================================
__global__ void __launch_bounds__(256)
k_attn(const float* __restrict__ q, const float* __restrict__ atom,
       const float* __restrict__ prot, const int* __restrict__ plen,
       float* __restrict__ h0) {
    __shared__ float Qs[48 * QS_STRIDE];          // 25.3 KB
    __shared__ float Ps[8 * 16 * QS_STRIDE];      // 67.6 KB: per-wave 16x128 P tile
    __shared__ float Wps[NT_TILES * 16];          // 1008 column maxima
    __shared__ float rowmax_s[48];
    __shared__ float Wc_s[NPB];
    __shared__ float red_s[256];
    __shared__ float scal_s[1];

    const int b    = blockIdx.x;
    const int tid  = threadIdx.x;
    const int lane = tid & 31;
    const int wv   = tid >> 5;
    const int n0   = b * NPB;
    const int len  = plen[b];

    // stage Q_b into LDS, zero-pad rows 40..47
    for (int i = tid; i < 48 * D_; i += 256) {
        int r = i >> 7, cc = i & (D_ - 1);
        Qs[r * QS_STRIDE + cc] = (r < NPB) ? q[(size_t)(n0 + r) * D_ + cc] : 0.0f;
    }
    if (tid < 48) rowmax_s[tid] = NEGV;
    __syncthreads();

    const int col  = lane & 15;
    const int half = lane >> 4;
    float* ptile = &Ps[wv * 16 * QS_STRIDE];      // wave-private tile buffer

    for (int nt = wv; nt < NT_TILES; nt += 8) {
        const int l0 = nt * 16;

        // ---- async-stage this wave's 16x128 P tile into LDS ----
        // WAR guard: previous tile's ds_loads must have drained
        asm volatile("s_wait_dscnt 0x0" ::: "memory");
        #pragma unroll
        for (int it = 0; it < 16; ++it) {
            const int cidx = it * 32 + lane;      // 16B chunk id, 0..511
            const int rr   = cidx >> 3;           // residue row in tile
            const int cw   = (cidx & 7) * 4;      // float offset within row
            int lg = l0 + rr; if (lg >= L_) lg = L_ - 1;   // clamp OOB addr
            const float* g = prot + ((size_t)b * L_ + lg) * D_ + cw;
            const unsigned lo = lds_off(&ptile[rr * QS_STRIDE + cw]);
            asm volatile("global_load_async_to_lds_b128 %0, %1, off"
                         :: "v"(lo), "v"(g) : "memory");
        }
        asm volatile("s_wait_asynccnt 0x0" ::: "memory");

        // L2-warm the next tile while we compute on this one
        {
            int ntn = nt + 8;
            if (ntn < NT_TILES) {
                int ln = ntn * 16 + col; if (ln >= L_) ln = L_ - 1;
                __builtin_prefetch(prot + ((size_t)b * L_ + ln) * D_, 0, 0);
            }
        }

        // ---- WMMA: 3 M-tiles x 32 k-steps, A and B both from LDS ----
        v8f c0 = {}, c1 = {}, c2 = {};
        #pragma unroll 4
        for (int ks = 0; ks < D_ / 4; ++ks) {
            const int kb = ks * 4 + half * 2;
            v2f bf = *(const v2f*)(&ptile[col * QS_STRIDE + kb]);   // B[k,n]=P[l0+n,k]
            v2f a0 = *(const v2f*)(&Qs[(0 * 16 + col) * QS_STRIDE + kb]);
            v2f a1 = *(const v2f*)(&Qs[(1 * 16 + col) * QS_STRIDE + kb]);
            v2f a2 = *(const v2f*)(&Qs[(2 * 16 + col) * QS_STRIDE + kb]);
            c0 = __builtin_amdgcn_wmma_f32_16x16x4_f32(false, a0, false, bf, (short)0, c0, false, false);
            c1 = __builtin_amdgcn_wmma_f32_16x16x4_f32(false, a1, false, bf, (short)0, c1, false, false);
            c2 = __builtin_amdgcn_wmma_f32_16x16x4_f32(false, a2, false, bf, (short)0, c2, false, false);
        }

        const int  l      = l0 + col;
        const bool lvalid = (l < len);
        float colmax = NEGV;
        auto proc = [&](const v8f& cc, int mtt) {
            #pragma unroll
            for (int r = 0; r < 8; ++r) {
                const int m = mtt * 16 + half * 8 + r;       // atom index in [0,48)
                float s = tanhf(cc[r]);
                if (!lvalid || m >= NPB) s = NEGV;
                colmax = fmaxf(colmax, s);
                // row max over the 16 lanes of this half-wave (same m, 16 l's)
                float v = s;
                v = fmaxf(v, __shfl_xor(v, 1));
                v = fmaxf(v, __shfl_xor(v, 2));
                v = fmaxf(v, __shfl_xor(v, 4));
                v = fmaxf(v, __shfl_xor(v, 8));
                if (col == 0 && m < NPB) atomicMaxF(&rowmax_s[m], v);
            }
        };
        proc(c0, 0); proc(c1, 1); proc(c2, 2);
        // combine the two halves (m 0..7+ vs 8..15+) for the same column l
        colmax = fmaxf(colmax, __shfl_xor(colmax, 16));
        if (half == 0) Wps[l0 + col] = colmax;               // Wp[b, l]
    }
    __syncthreads();

    // ---- atom-side pooling: aa = exp(rowmax)/sum; atom_pool = sum aa*atom ----
    if (tid < NPB) Wc_s[tid] = expf(rowmax_s[tid]);
    __syncthreads();
    if (tid == 0) {
        float s = 0.0f;
        for (int i = 0; i < NPB; ++i) s += Wc_s[i];
        scal_s[0] = s;
    }
    __syncthreads();
    if (tid < D_) {
        const float inv = 1.0f / scal_s[0];
        float acc = 0.0f;
        for (int n = 0; n < NPB; ++n)
            acc += Wc_s[n] * atom[(size_t)(n0 + n) * D_ + tid];
        h0[(size_t)b * (2 * D_) + tid] = acc * inv;
    }

    // ---- protein-side pooling: softmax(Wp) over L, then ap @ P_b ----
    float lm = NEGV;
    for (int i = tid; i < NT_TILES * 16; i += 256) lm = fmaxf(lm, Wps[i]);
    red_s[tid] = lm;
    __syncthreads();
    for (int s = 128; s > 0; s >>= 1) {
        if (tid < s) red_s[tid] = fmaxf(red_s[tid], red_s[tid + s]);
        __syncthreads();
    }
    const float maxv = red_s[0];
    __syncthreads();
    float ls = 0.0f;
    for (int i = tid; i < NT_TILES * 16; i += 256) {
        float e = expf(Wps[i] - maxv);                       // masked -> exp(-inf)=0
        Wps[i] = e;
        ls += e;
    }
    red_s[tid] = ls;
    __syncthreads();
    for (int s = 128; s > 0; s >>= 1) {
        if (tid < s) red_s[tid] += red_s[tid + s];
        __syncthreads();
    }
    const float sumv = red_s[0];
    __syncthreads();
    if (tid < D_) {
        float acc = 0.0f;
        const float* pb = prot + (size_t)b * L_ * D_;
        for (int l = 0; l < L_; ++l)
            acc += Wps[l] * pb[(size_t)l * D_ + tid];
        h0[(size_t)b * (2 * D_) + D_ + tid] = acc / sumv;
    }
}

// ============================================================
// Kernel 3: MLP layers, Y = act(X @ W + bias), WMMA-tiled
// ============================================================
template <int K, int NCOLS>
__global__ void __launch_bounds__(256)
k_mlp(const float* __restrict__ X, const float* __restrict__ W,
      const float* __restrict__ bias, float* __restrict__ Y, int relu) {
    const int lane = threadIdx.x & 31;
    const int wid  = blockIdx.x * 8 + (threadIdx.x >> 5);
    const int nt_n = NCOLS / 16;
    const int mt   = wid / nt_n;
    const int nt   = wid % nt_n;
    const int col  = lane & 15;
    const int half = lane >> 4;

    const float* xrow = X + (size_t)(mt * 16 + col) * K;
    v8f c = {};
    #pragma unroll 4
    for (int ks = 0; ks < K / 4; ++ks) {
        const int kb = ks * 4 + half * 2;
        v2f a = *(const v2f*)(xrow + kb);
        v2f b;
        b.x = W[(size_t)kb * NCOLS + nt * 16 + col];
        b.y = W[(size_t)(kb + 1) * NCOLS + nt * 16 + col];
        c = __builtin_amdgcn_wmma_f32_16x16x4_f32(false, a, false, b,
                                                  (short)0, c, false, false);
    }
    const float bv = bias[nt * 16 + col];
    float* yo = Y + (size_t)(mt * 16 + half * 8) * NCOLS + nt * 16 + col;
    #pragma unroll
    for (int r = 0; r < 8; ++r) {
        float v = c[r] + bv;
        if (relu) v = fmaxf(v, 0.0f);
        yo[(size_t)r * NCOLS] = v;
    }
}

// final [64,256] @ [256,1] + bo
__global__ void k_out(const float* __restrict__ h2, const float* __restrict__ Wo,
                      const float* __restrict__ bo, float* __restrict__ out) {
    int m = threadIdx.x;
    if (m < B_) {
        float acc = 0.0f;
        const float* hr = h2 + (size_t)m * H2_;
        for (int d = 0; d < H2_; ++d) acc += hr[d] * Wo[d];
        out[m] = acc + bo[0];
    }
}

extern "C" void kernel_launch(void* const* d_in, const int* in_sizes, int n_in,
                              void* d_out, int out_size, void* d_ws, size_t ws_size,
                              hipStream_t stream) {
    (void)in_sizes; (void)n_in; (void)out_size; (void)ws_size;
    const float* atom = (const float*)d_in[0];   // [2560,128]
    const float* prot = (const float*)d_in[1];   // [64,1000,128]
    /* d_in[2] = atom_splits: fixed layout repeat(arange(64),40) -> b = n/40 */
    const int*   plen = (const int*)d_in[3];     // [64]
    const float* Wa   = (const float*)d_in[4];   // [128,128]
    const float* W1   = (const float*)d_in[5];   // [256,512]
    const float* b1   = (const float*)d_in[6];   // [512]
    const float* W2   = (const float*)d_in[7];   // [512,256]
    const float* b2   = (const float*)d_in[8];   // [256]
    const float* Wo   = (const float*)d_in[9];   // [256,1]
    const float* bo   = (const float*)d_in[10];  // [1]
    float* out = (float*)d_out;

    float* ws = (float*)d_ws;
    float* q  = ws;                              // 2560*128
    float* h0 = q  + (size_t)N_ * D_;            // 64*256
    float* h1 = h0 + (size_t)B_ * 2 * D_;        // 64*512
    float* h2 = h1 + (size_t)B_ * H1_;           // 64*256

    k_q_gemm<<<dim3(N_ / 16), dim3(256), 0, stream>>>(atom, Wa, q);
    k_attn<<<dim3(B_), dim3(256), 0, stream>>>(q, atom, prot, plen, h0);
    k_mlp<2 * D_, H1_><<<dim3((4 * (H1_ / 16)) / 8), dim3(256), 0, stream>>>(h0, W1, b1, h1, 1);
    k_mlp<H1_, H2_><<<dim3((4 * (H2_ / 16)) / 8), dim3(256), 0, stream>>>(h1, W2, b2, h2, 1);
    k_out<<<dim3(1), dim3(64), 0, stream>>>(h2, Wo, bo, out);
}